// Encoder_27496380629729
// MI455X (gfx1250) — compile-verified
//
#include <hip/hip_runtime.h>

typedef __attribute__((ext_vector_type(2))) float v2f;
typedef __attribute__((ext_vector_type(8))) float v8f;

// ---------------- utility kernels ----------------

__global__ void zero_f32(float* __restrict__ p, long long n) {
  long long i = (long long)blockIdx.x * blockDim.x + threadIdx.x;
  long long stride = (long long)gridDim.x * blockDim.x;
  for (; i < n; i += stride) p[i] = 0.0f;
}

__global__ void deg_count(const long long* __restrict__ dst,
                          float* __restrict__ cnt, long long E) {
  long long i = (long long)blockIdx.x * blockDim.x + threadIdx.x;
  long long stride = (long long)gridDim.x * blockDim.x;
  for (; i < E; i += stride) {
    atomicAdd(&cnt[(int)dst[i]], 1.0f);
  }
}

__global__ void make_dinv(const float* __restrict__ cnt,
                          float* __restrict__ dinv, int n) {
  int i = blockIdx.x * blockDim.x + threadIdx.x;
  if (i < n) {
    // self-loop contributes +1; degree always > 0
    dinv[i] = rsqrtf(cnt[i] + 1.0f);
  }
}

// ---------------- fp32 WMMA GEMM: Out[nrows,N] = X[nrows,K] @ W[K,N] --------
// K, N compile-time: all LDS / global load offsets become immediates, the
// K-loop fully unrolls into 32 chained V_WMMA_F32_16X16X4_F32 ops.
// One 16-row slab per block; each of the 2*N/32 waves owns one 16-col tile.
//
// Fragment layouts per CDNA5 ISA 7.12.2:
//  A (16x4 f32, 2 VGPRs): lanes0-15 -> K=0,1 ; lanes16-31 -> K=2,3 ; M=lane&15
//  B (4x16 f32, 2 VGPRs): mirrored striping, N=lane&15
//  C/D (16x16 f32, 8 VGPRs): vgpr r: M = r + 8*(lane>>4), N = lane&15

template <int K, int N>
__global__ void gemm_wmma_f32(const float* __restrict__ X,
                              const float* __restrict__ W,
                              float* __restrict__ Out, int nrows) {
  constexpr int LDK = K + 4;                // 4-float skew -> conflict-free cols
  __shared__ float ldsX[16 * LDK];
  const int rbase = blockIdx.x << 4;
  const int tid   = threadIdx.x;

  // --- cooperative, coalesced stage of the 16xK X slab into LDS (float4) ---
  constexpr int KV = K / 4;                 // float4s per row
  constexpr int NV = 16 * KV;
#pragma unroll
  for (int v = tid; v < NV; v += 2 * N) {   // blockDim == 2*N
    const int r = v / KV;
    const int c = v - r * KV;
    float4 val = {0.f, 0.f, 0.f, 0.f};
    const int row = rbase + r;
    if (row < nrows)
      val = ((const float4*)(X + (size_t)row * K))[c];
    ((float4*)(ldsX + r * LDK))[c] = val;   // (K+4)*4B rows stay 16B aligned
  }
  __syncthreads();

  const int lane  = tid & 31;
  const int cbase = (tid >> 5) << 4;        // wave id -> 16-col tile (< N by construction)
  const int m     = lane & 15;              // A row / B,D col within tile
  const int kq    = (lane >> 4) << 1;       // K sub-offset per half-wave

  const float* ap0 = ldsX + m * LDK + kq;
  const float* bp0 = W + (size_t)kq * N + cbase + m;

  v8f acc = {};
#pragma unroll
  for (int k0 = 0; k0 < K; k0 += 4) {
    v2f a, b;
    a.x = ap0[k0];                          // immediate LDS offsets
    a.y = ap0[k0 + 1];
    b.x = bp0[k0 * N];                      // immediate global offsets
    b.y = bp0[k0 * N + N];
    // (neg_a, A, neg_b, B, c_mod, C, reuse_a, reuse_b)
    acc = __builtin_amdgcn_wmma_f32_16x16x4_f32(false, a, false, b,
                                                (short)0, acc, false, false);
  }

  const int rowoff = rbase + ((lane >> 4) << 3);
  const int col    = cbase + (lane & 15);
#pragma unroll
  for (int r = 0; r < 8; ++r) {
    const int row = rowoff + r;
    if (row < nrows)
      Out[(size_t)row * N + col] = acc[r];
  }
}

// ---------------- edge scatter: agg[dst] += H[src] * dinv[src]*dinv[dst] ----
// One thread per (edge, 4 features): float4 gather (global_load_b128) +
// 4 fire-and-forget global_atomic_add_f32. agg (<=25 MB) is L2-resident.

__global__ void edge_scatter(const float* __restrict__ H,
                             const long long* __restrict__ src,
                             const long long* __restrict__ dst,
                             const float* __restrict__ dinv,
                             float* __restrict__ agg,
                             long long E, int logFV) {  // logFV = log2(F/4)
  const long long total = E << logFV;
  const int fmask = (1 << logFV) - 1;
  long long t = (long long)blockIdx.x * blockDim.x + threadIdx.x;
  const long long stride = (long long)gridDim.x * blockDim.x;
  for (; t < total; t += stride) {
    const long long e = t >> logFV;
    const int f4 = (int)(t & fmask);
    const int s  = (int)src[e];
    const int d  = (int)dst[e];
    const float nrm = dinv[s] * dinv[d];
    const float4 h = ((const float4*)(H + ((size_t)s << (logFV + 2))))[f4];
    float* ap = agg + ((size_t)d << (logFV + 2)) + 4 * (size_t)f4;
    atomicAdd(ap + 0, h.x * nrm);
    atomicAdd(ap + 1, h.y * nrm);
    atomicAdd(ap + 2, h.z * nrm);
    atomicAdd(ap + 3, h.w * nrm);
  }
}

// ------- finalize: out = relu(agg + H*dinv^2 + bias), float4, in-place OK ---

__global__ void finalize_relu(const float* __restrict__ H,
                              const float* __restrict__ dinv,
                              const float* __restrict__ bias,
                              const float* __restrict__ aggin,
                              float* __restrict__ out,
                              long long n4, int logFV) {  // n4 = N*F/4
  const int fmask = (1 << logFV) - 1;
  long long t = (long long)blockIdx.x * blockDim.x + threadIdx.x;
  const long long stride = (long long)gridDim.x * blockDim.x;
  for (; t < n4; t += stride) {
    const int i  = (int)(t >> logFV);
    const int f4 = (int)(t & fmask);
    const float di = dinv[i];
    const float d2 = di * di;
    const float4 a = ((const float4*)aggin)[t];
    const float4 h = ((const float4*)H)[t];
    const float4 b = ((const float4*)bias)[f4];
    float4 r;
    r.x = fmaxf(a.x + h.x * d2 + b.x, 0.0f);
    r.y = fmaxf(a.y + h.y * d2 + b.y, 0.0f);
    r.z = fmaxf(a.z + h.z * d2 + b.z, 0.0f);
    r.w = fmaxf(a.w + h.w * d2 + b.w, 0.0f);
    ((float4*)out)[t] = r;
  }
}

// ---------------- driver ----------------

extern "C" void kernel_launch(void* const* d_in, const int* in_sizes, int n_in,
                              void* d_out, int out_size, void* d_ws, size_t ws_size,
                              hipStream_t stream) {
  (void)n_in; (void)out_size; (void)ws_size;
  constexpr int DIN = 128, HID = 128, DOUT = 64;
  constexpr int LOG_HV = 5;   // log2(HID/4)
  constexpr int LOG_OV = 4;   // log2(DOUT/4)

  const float*     x   = (const float*)d_in[0];
  const long long* ei  = (const long long*)d_in[1];
  const float*     W1  = (const float*)d_in[2];
  const float*     b1  = (const float*)d_in[3];
  const float*     W2  = (const float*)d_in[4];
  const float*     b2  = (const float*)d_in[5];
  float*           out = (float*)d_out;

  const int       N = in_sizes[0] / DIN;     // 50000
  const long long E = in_sizes[1] / 2;       // 800000
  const long long* src = ei;
  const long long* dst = ei + E;

  // workspace partition (floats)
  float* ws   = (float*)d_ws;
  float* cnt  = ws;                          // N
  float* dinv = cnt  + N;                    // N
  float* h1   = dinv + N;                    // N*HID
  float* agg1 = h1   + (size_t)N * HID;      // N*HID (finalized in place)
  float* h2   = agg1 + (size_t)N * HID;      // N*DOUT
  float* agg2 = h2   + (size_t)N * DOUT;     // N*DOUT

  const long long nh1 = (long long)N * HID;
  const long long nh2 = (long long)N * DOUT;

  // 1) degree -> dinv
  zero_f32<<<(N + 255) / 256, 256, 0, stream>>>(cnt, N);
  deg_count<<<(int)((E + 255) / 256), 256, 0, stream>>>(dst, cnt, E);
  make_dinv<<<(N + 255) / 256, 256, 0, stream>>>(cnt, dinv, N);

  // 2) layer 1: h1 = x @ W1 (fp32 WMMA, 8 waves/block)
  gemm_wmma_f32<DIN, HID><<<(N + 15) / 16, 2 * HID, 0, stream>>>(x, W1, h1, N);
  zero_f32<<<(int)((nh1 + 255) / 256), 256, 0, stream>>>(agg1, nh1);
  edge_scatter<<<(int)(((E << LOG_HV) + 255) / 256), 256, 0, stream>>>(
      h1, src, dst, dinv, agg1, E, LOG_HV);
  finalize_relu<<<(int)((nh1 / 4 + 255) / 256), 256, 0, stream>>>(
      h1, dinv, b1, agg1, agg1, nh1 / 4, LOG_HV);   // agg1 now holds relu'd h

  // 3) layer 2: h2 = agg1 @ W2 (fp32 WMMA, 4 waves/block)
  gemm_wmma_f32<HID, DOUT><<<(N + 15) / 16, 2 * DOUT, 0, stream>>>(agg1, W2, h2, N);
  zero_f32<<<(int)((nh2 + 255) / 256), 256, 0, stream>>>(agg2, nh2);
  edge_scatter<<<(int)(((E << LOG_OV) + 255) / 256), 256, 0, stream>>>(
      h2, src, dst, dinv, agg2, E, LOG_OV);
  finalize_relu<<<(int)((nh2 / 4 + 255) / 256), 256, 0, stream>>>(
      h2, dinv, b2, agg2, out, nh2 / 4, LOG_OV);
}